// MRFStyleLoss_90761248899172
// MI455X (gfx1250) — compile-verified
//
#include <hip/hip_runtime.h>

// ---------------------------------------------------------------------------
// MRF style loss for MI455X (gfx1250, wave32, WMMA).
// Core: 16129x16129x27 f32 GEMM reduced on-the-fly to per-row argmax using
// V_WMMA_F32_16X16X4_F32. Style fragments are pre-scaled by 1/||s||, the
// partial last style tile is peeled, and each wave runs two independent
// accumulator chains (2 syn tiles) to break the WMMA RAW chain and halve
// B-fragment traffic.
// ---------------------------------------------------------------------------

typedef float v2f __attribute__((ext_vector_type(2)));
typedef float v8f __attribute__((ext_vector_type(8)));

#define HH 256
#define WW 256
#define CC 3
#define PS 3
#define STR 2
#define NSIDE 127                    // (256-3)/2 + 1
#define NP (NSIDE * NSIDE)           // 16129 patches per map
#define KDIM 27                      // 3*3*3
#define KP 28                        // padded to multiple of 4
#define NKSTEP 7                     // KP / 4
#define NTILES ((NP + 15) / 16)      // 1009 tiles of 16 rows
#define NP_PAD (NTILES * 16)         // 16144
#define WPB 8                        // waves per block
#define QT_PER_WAVE 2                // syn tiles per wave
#define NWAVES_NEEDED ((NTILES + QT_PER_WAVE - 1) / QT_PER_WAVE)   // 505
#define NBLOCKS ((NWAVES_NEEDED + WPB - 1) / WPB)                  // 64
#define QTILES_ALLOC (NBLOCKS * WPB * QT_PER_WAVE)                 // 1024
#define FRAG_FLOATS (NKSTEP * 64)    // floats per fragment tile = 448
#define NEG_INF (-3.4e38f)

// ---------------------------------------------------------------- zero init
__global__ void mrf_zero(float* __restrict__ p, size_t n) {
    size_t i = (size_t)blockIdx.x * blockDim.x + threadIdx.x;
    size_t stride = (size_t)gridDim.x * blockDim.x;
    for (; i < n; i += stride) p[i] = 0.0f;
}

// ------------------------------------------------------- patch extraction
// Writes row-major padded rows [NP_PAD x 28] (raw values, used by the MSE
// gather) and the WMMA-fragment layout. Style fragments are pre-scaled by
// the inverse Frobenius norm so the GEMM emits normalized scores directly.
__global__ void mrf_extract(const float* __restrict__ img,
                            float* __restrict__ row,
                            float* __restrict__ frag,
                            int isStyle) {
    int p = blockIdx.x * blockDim.x + threadIdx.x;
    if (p >= NP) return;
    int i = p / NSIDE;
    int j = p - i * NSIDE;
    int r0 = i * STR;
    int c0 = j * STR;

    float vals[KDIM];
    float ss = 0.0f;
#pragma unroll
    for (int ch = 0; ch < CC; ++ch)
#pragma unroll
        for (int dy = 0; dy < PS; ++dy)
#pragma unroll
            for (int dx = 0; dx < PS; ++dx) {
                int k = ch * 9 + dy * 3 + dx;
                float v = img[ch * (HH * WW) + (r0 + dy) * WW + (c0 + dx)];
                vals[k] = v;
                ss += v * v;
            }

    float* rp = row + (size_t)p * KP;
#pragma unroll
    for (int k = 0; k < KDIM; ++k) rp[k] = vals[k];
    rp[KDIM] = 0.0f;

    float inv = isStyle ? (1.0f / sqrtf(ss)) : 1.0f;

    // ISA 32-bit 16x4 fragment striping:
    //   lane = (row&15) + (k%4>=2 ? 16 : 0), VGPR(sub) = k&1
    int tile = p >> 4;
    int lsub = p & 15;
    float* fp = frag + (size_t)tile * FRAG_FLOATS;
#pragma unroll
    for (int k = 0; k < KDIM; ++k) {
        int ks   = k >> 2;
        int kk   = k & 3;
        int lane = lsub + ((kk >= 2) ? 16 : 0);
        int sub  = kk & 1;
        fp[ks * 64 + lane * 2 + sub] = vals[k] * inv;
    }
}

// -------------------------------------------- fused GEMM + row-wise argmax
__global__ __launch_bounds__(256) void mrf_argmax(
        const float* __restrict__ qfrag, const float* __restrict__ sfrag,
        int* __restrict__ match) {
    const int lane = threadIdx.x & 31;
    const int w = blockIdx.x * WPB + (threadIdx.x >> 5);   // wave id
    const int qt0 = 2 * w;                                 // two syn tiles

    // Preload 2 x 7 A fragments (32 syn rows) — constant over style loop.
    const float* qf0 = qfrag + (size_t)qt0 * FRAG_FLOATS + lane * 2;
    const float* qf1 = qf0 + FRAG_FLOATS;
    v2f a0[NKSTEP], a1[NKSTEP];
#pragma unroll
    for (int ks = 0; ks < NKSTEP; ++ks) {
        a0[ks] = *(const v2f*)(qf0 + ks * 64);
        a1[ks] = *(const v2f*)(qf1 + ks * 64);
    }

    float bV0[8], bV1[8];
    int bI0[8], bI1[8];
#pragma unroll
    for (int jj = 0; jj < 8; ++jj) {
        bV0[jj] = NEG_INF; bI0[jj] = 0;
        bV1[jj] = NEG_INF; bI1[jj] = 0;
    }

    const int nsub = lane & 15;  // D layout: lane holds column N = lane&15
    const float* sf = sfrag + lane * 2;

    // ---- main loop: full style tiles, no masking, two independent chains
    for (int st = 0; st < NTILES - 1; ++st) {
        v8f acc0 = {0.f, 0.f, 0.f, 0.f, 0.f, 0.f, 0.f, 0.f};
        v8f acc1 = {0.f, 0.f, 0.f, 0.f, 0.f, 0.f, 0.f, 0.f};
#pragma unroll
        for (int ks = 0; ks < NKSTEP; ++ks) {
            v2f b = *(const v2f*)(sf + ks * 64);
            acc0 = __builtin_amdgcn_wmma_f32_16x16x4_f32(
                false, a0[ks], false, b, (short)0, acc0, false, false);
            acc1 = __builtin_amdgcn_wmma_f32_16x16x4_f32(
                false, a1[ks], false, b, (short)0, acc1, false, false);
        }
        sf += FRAG_FLOATS;
        int col = st * 16 + nsub;
#pragma unroll
        for (int jj = 0; jj < 8; ++jj) {
            float v0 = acc0[jj];
            float v1 = acc1[jj];
            if (v0 > bV0[jj]) { bV0[jj] = v0; bI0[jj] = col; }
            if (v1 > bV1[jj]) { bV1[jj] = v1; bI1[jj] = col; }
        }
    }

    // ---- epilogue: last (partial) style tile with column masking
    {
        v8f acc0 = {0.f, 0.f, 0.f, 0.f, 0.f, 0.f, 0.f, 0.f};
        v8f acc1 = {0.f, 0.f, 0.f, 0.f, 0.f, 0.f, 0.f, 0.f};
#pragma unroll
        for (int ks = 0; ks < NKSTEP; ++ks) {
            v2f b = *(const v2f*)(sf + ks * 64);
            acc0 = __builtin_amdgcn_wmma_f32_16x16x4_f32(
                false, a0[ks], false, b, (short)0, acc0, false, false);
            acc1 = __builtin_amdgcn_wmma_f32_16x16x4_f32(
                false, a1[ks], false, b, (short)0, acc1, false, false);
        }
        int col = (NTILES - 1) * 16 + nsub;
        bool valid = (col < NP);
#pragma unroll
        for (int jj = 0; jj < 8; ++jj) {
            float v0 = valid ? acc0[jj] : NEG_INF;
            float v1 = valid ? acc1[jj] : NEG_INF;
            if (v0 > bV0[jj]) { bV0[jj] = v0; bI0[jj] = col; }
            if (v1 > bV1[jj]) { bV1[jj] = v1; bI1[jj] = col; }
        }
    }

    // ---- cross-lane argmax within each 16-lane half (xor masks stay in-half)
#pragma unroll
    for (int m = 8; m >= 1; m >>= 1) {
#pragma unroll
        for (int jj = 0; jj < 8; ++jj) {
            float ov0 = __shfl_xor(bV0[jj], m, 32);
            int oi0 = __shfl_xor(bI0[jj], m, 32);
            if (ov0 > bV0[jj] || (ov0 == bV0[jj] && oi0 < bI0[jj])) {
                bV0[jj] = ov0; bI0[jj] = oi0;
            }
            float ov1 = __shfl_xor(bV1[jj], m, 32);
            int oi1 = __shfl_xor(bI1[jj], m, 32);
            if (ov1 > bV1[jj] || (ov1 == bV1[jj] && oi1 < bI1[jj])) {
                bV1[jj] = ov1; bI1[jj] = oi1;
            }
        }
    }

    // VGPR jj: lanes 0-15 -> row jj, lanes 16-31 -> row jj+8.
    if (nsub == 0) {
        int half8 = (lane >> 4) * 8;
        int rowBase0 = qt0 * 16 + half8;
        int rowBase1 = (qt0 + 1) * 16 + half8;
#pragma unroll
        for (int jj = 0; jj < 8; ++jj) {
            int r0 = rowBase0 + jj;
            int r1 = rowBase1 + jj;
            if (r0 < NP) match[r0] = bI0[jj];
            if (r1 < NP) match[r1] = bI1[jj];
        }
    }
}

// -------------------------------------------------- per-patch MSE gather
__global__ void mrf_mse(const float* __restrict__ qrow,
                        const float* __restrict__ srow,
                        const int* __restrict__ match,
                        float* __restrict__ mse) {
    int p = blockIdx.x * blockDim.x + threadIdx.x;
    if (p >= NP) return;
    int m = match[p];
    const float* q = qrow + (size_t)p * KP;
    const float* s = srow + (size_t)m * KP;
    float acc = 0.0f;
#pragma unroll
    for (int k = 0; k < KDIM; ++k) {
        float d = q[k] - s[k];
        acc += d * d;
    }
    mse[p] = acc * (1.0f / (float)KDIM);
}

// ----------------------------------------------- deterministic reduction
__global__ __launch_bounds__(256) void mrf_reduce(const float* __restrict__ mse,
                                                  float* __restrict__ out) {
    __shared__ float sm[256];
    float acc = 0.0f;
    for (int i = threadIdx.x; i < NP; i += 256) acc += mse[i];
    sm[threadIdx.x] = acc;
    __syncthreads();
    for (int s = 128; s > 0; s >>= 1) {
        if ((int)threadIdx.x < s) sm[threadIdx.x] += sm[threadIdx.x + s];
        __syncthreads();
    }
    if (threadIdx.x == 0) out[0] = sm[0] * (1.0f / (float)NP);
}

// ---------------------------------------------------------------------------
extern "C" void kernel_launch(void* const* d_in, const int* in_sizes, int n_in,
                              void* d_out, int out_size, void* d_ws, size_t ws_size,
                              hipStream_t stream) {
    (void)in_sizes; (void)n_in; (void)out_size; (void)ws_size;
    const float* syn = (const float*)d_in[0];
    const float* sty = (const float*)d_in[1];
    float* out = (float*)d_out;

    float* ws = (float*)d_ws;
    size_t o = 0;
    float* qrow  = ws + o;  o += (size_t)NP_PAD * KP;                 // 452032
    float* srow  = ws + o;  o += (size_t)NP_PAD * KP;                 // 452032
    float* qfrag = ws + o;  o += (size_t)QTILES_ALLOC * FRAG_FLOATS;  // 458752
    float* sfrag = ws + o;  o += (size_t)NTILES * FRAG_FLOATS;        // 452032
    float* mse   = ws + o;  o += NP_PAD;
    int* match   = (int*)(ws + o);

    // Zero fragment arrays (padding rows must contribute zero to WMMA).
    size_t fragTotal = (size_t)QTILES_ALLOC * FRAG_FLOATS +
                       (size_t)NTILES * FRAG_FLOATS;
    mrf_zero<<<512, 256, 0, stream>>>(qfrag, fragTotal);

    int exBlocks = (NP + 255) / 256;
    mrf_extract<<<exBlocks, 256, 0, stream>>>(syn, qrow, qfrag, 0);
    mrf_extract<<<exBlocks, 256, 0, stream>>>(sty, srow, sfrag, 1);

    mrf_argmax<<<NBLOCKS, WPB * 32, 0, stream>>>(qfrag, sfrag, match);

    mrf_mse<<<exBlocks, 256, 0, stream>>>(qrow, srow, match, mse);
    mrf_reduce<<<1, 256, 0, stream>>>(mse, out);
}